// AxialUNet_49813030699176
// MI455X (gfx1250) — compile-verified
//
#include <hip/hip_runtime.h>
#include <hip/hip_bf16.h>
#include <math.h>

// ============================================================================
// AxialUNet forward for MI455X (gfx1250, wave32).
// All matrix contractions -> V_WMMA_F32_16X16X4_F32 (fp32, exact vs reference;
// the net is HBM/L2-bandwidth bound so fp32 WMMA is the right precision).
// GEMM: one wave computes a 16x64 macro-tile (4 16x16 WMMA tiles sharing one
// A fragment -> 4x less strided A traffic, 4 independent accumulators for
// back-to-back WMMA issue). transB is a template parameter (no runtime layout
// branch); edge tiles use clamped loads + multiplicative masks so the
// compiler emits unconditional straight-line loads (no exec diamonds).
// ============================================================================

typedef float v2f __attribute__((ext_vector_type(2)));
typedef float v8f __attribute__((ext_vector_type(8)));

#define TPB 256   // 8 waves of 32

#define WMMA_F32(a, b, c) \
  __builtin_amdgcn_wmma_f32_16x16x4_f32(false, (a), false, (b), (short)0, (c), false, false)

// ---------------------------------------------------------------------------
// Generic strided-batched GEMM.
//   C[g] = alpha * A[g](MxK) * op(B[g]) (+bias[n]) (+relu)
//   op(B) = B (KxN, row stride ldb)            (TRANSB = 0)
//         = B^T with B stored NxK, row stride ldb (TRANSB = 1)
// Fragment layouts per CDNA5 ISA 7.12.2 (32-bit A 16x4 / B 4x16 / C 16x16).
// Grid: tiles of 16 rows x 64 cols; tilesN4 = ceil(N/64).
// ---------------------------------------------------------------------------
template <int TRANSB>
__global__ __launch_bounds__(TPB) void gemm_wmma_f32(
    const float* __restrict__ A, const float* __restrict__ B,
    float* __restrict__ C, const float* __restrict__ bias,
    int M, int N, int K, int lda, int ldb, int ldc,
    long long sA, long long sB, long long sC,
    float alpha, int relu, int tilesN4)
{
  const int lane = threadIdx.x & 31;
  const int wave = threadIdx.x >> 5;
  const long long tile = (long long)blockIdx.x * (TPB / 32) + wave;
  const int tm  = (int)(tile / tilesN4);
  const int tn4 = (int)(tile % tilesN4);
  if (tm * 16 >= M) return;

  const int g = blockIdx.y;
  A += (long long)g * sA;
  B += (long long)g * sB;
  C += (long long)g * sC;

  const int half = lane >> 4;     // 0: K pair {0,1}; 1: K pair {2,3}
  const int l16  = lane & 15;
  const int row  = tm * 16 + l16; // A row this lane supplies
  const int n0   = tn4 * 64;

  const bool fast = (tm * 16 + 16 <= M) && (n0 + 64 <= N) && ((K & 3) == 0);

  if (fast) {
    v8f acc0 = {0.f,0.f,0.f,0.f,0.f,0.f,0.f,0.f};
    v8f acc1 = acc0, acc2 = acc0, acc3 = acc0;
    const float* Ap = A + (long long)row * lda + half * 2;

    if (!TRANSB) {
      const float* Bp = B + (long long)(half * 2) * ldb + n0 + l16;
#pragma unroll 2
      for (int k = 0; k < K; k += 4) {
        const v2f a = *(const v2f*)(Ap + k);
        const float* br = Bp + (long long)k * ldb;
        v2f b0, b1, b2, b3;
        b0.x = br[0];  b0.y = br[ldb];
        b1.x = br[16]; b1.y = br[ldb + 16];
        b2.x = br[32]; b2.y = br[ldb + 32];
        b3.x = br[48]; b3.y = br[ldb + 48];
        acc0 = WMMA_F32(a, b0, acc0);
        acc1 = WMMA_F32(a, b1, acc1);
        acc2 = WMMA_F32(a, b2, acc2);
        acc3 = WMMA_F32(a, b3, acc3);
      }
    } else {
      const float* Bp = B + half * 2;
      const long long c0 = (long long)(n0 + l16) * ldb;
#pragma unroll 2
      for (int k = 0; k < K; k += 4) {
        const v2f a  = *(const v2f*)(Ap + k);
        const v2f b0 = *(const v2f*)(Bp + c0 + k);
        const v2f b1 = *(const v2f*)(Bp + c0 + 16LL * ldb + k);
        const v2f b2 = *(const v2f*)(Bp + c0 + 32LL * ldb + k);
        const v2f b3 = *(const v2f*)(Bp + c0 + 48LL * ldb + k);
        acc0 = WMMA_F32(a, b0, acc0);
        acc1 = WMMA_F32(a, b1, acc1);
        acc2 = WMMA_F32(a, b2, acc2);
        acc3 = WMMA_F32(a, b3, acc3);
      }
    }

    // C layout: vgpr r -> row tm*16 + half*8 + r, col = lane-group column.
    v8f accs[4] = {acc0, acc1, acc2, acc3};
#pragma unroll
    for (int t = 0; t < 4; ++t) {
      const int n = n0 + t * 16 + l16;
      const float bi = bias ? bias[n] : 0.f;
      float* cp = C + (long long)(tm * 16 + half * 8) * ldc + n;
#pragma unroll
      for (int r = 0; r < 8; ++r) {
        float v = accs[t][r] * alpha + bi;
        if (relu) v = fmaxf(v, 0.f);
        cp[(long long)r * ldc] = v;
      }
    }
    return;
  }

  // ------- generic edge path: clamped addresses + multiplicative masks -----
  // (0.f * x cannot be folded, so loads stay unconditional & branch-free;
  //  clamped elements are real in-bounds matrix data, so 0*finite == 0.)
  const int rc      = min(row, M - 1);
  const float rmask = (row < M) ? 1.f : 0.f;
  const float* Ar   = A + (long long)rc * lda;

#pragma unroll 1
  for (int t = 0; t < 4; ++t) {
    const int cn = n0 + t * 16;
    if (cn >= N) break;
    const int col     = cn + l16;
    const int cc      = min(col, N - 1);
    const float cmask = (col < N) ? 1.f : 0.f;
    const float* Bc   = TRANSB ? (B + (long long)cc * ldb) : (B + cc);

    v8f acc = {0.f,0.f,0.f,0.f,0.f,0.f,0.f,0.f};
    for (int k0 = 0; k0 < K; k0 += 4) {
      const int ka  = k0 + half * 2;
      const int k0c = min(ka, K - 1);
      const int k1c = min(ka + 1, K - 1);
      const float km0 = (ka     < K) ? 1.f : 0.f;
      const float km1 = (ka + 1 < K) ? 1.f : 0.f;
      v2f a, b;
      a.x = Ar[k0c] * (rmask * km0);
      a.y = Ar[k1c] * (rmask * km1);
      if (!TRANSB) {
        b.x = Bc[(long long)k0c * ldb] * (cmask * km0);
        b.y = Bc[(long long)k1c * ldb] * (cmask * km1);
      } else {
        b.x = Bc[k0c] * (cmask * km0);
        b.y = Bc[k1c] * (cmask * km1);
      }
      acc = WMMA_F32(a, b, acc);
    }
#pragma unroll
    for (int r = 0; r < 8; ++r) {
      const int m = tm * 16 + half * 8 + r;
      if (m < M && col < N) {
        float v = acc[r] * alpha;
        if (bias) v += bias[col];
        if (relu) v = fmaxf(v, 0.f);
        C[(long long)m * ldc + col] = v;
      }
    }
  }
}

// ---------------------------------------------------------------------------
// Row softmax: one wave32 per row of length T.
// ---------------------------------------------------------------------------
__global__ __launch_bounds__(TPB) void softmax_rows(float* __restrict__ x,
                                                    long long rows, int T)
{
  const long long row = (long long)blockIdx.x * (TPB / 32) + (threadIdx.x >> 5);
  if (row >= rows) return;
  float* p = x + row * (long long)T;
  const int lane = threadIdx.x & 31;

  float mx = -3.4e38f;
  for (int i = lane; i < T; i += 32) mx = fmaxf(mx, p[i]);
  for (int o = 16; o; o >>= 1) mx = fmaxf(mx, __shfl_xor(mx, o, 32));

  float s = 0.f;
  for (int i = lane; i < T; i += 32) { float e = __expf(p[i] - mx); p[i] = e; s += e; }
  for (int o = 16; o; o >>= 1) s += __shfl_xor(s, o, 32);

  const float inv = 1.0f / s;
  for (int i = lane; i < T; i += 32) p[i] *= inv;
}

// ---------------------------------------------------------------------------
// Pack (B,C,H,W) -> channel-last token matrix for one attention axis.
// axis 0 (attend over H): row = (b*W + w)*H + h ; axis 1: row = (b*H + h)*W + w
// ---------------------------------------------------------------------------
__global__ __launch_bounds__(TPB) void pack_axis(
    const float* __restrict__ x, float* __restrict__ Xa,
    int Bv, int C, int H, int W, int axis, long long total)
{
  long long i = (long long)blockIdx.x * TPB + threadIdx.x;
  if (i >= total) return;
  const int c = (int)(i % C);
  long long r = i / C;
  const int Ad = axis == 0 ? W : H;
  const int Td = axis == 0 ? H : W;
  const int t = (int)(r % Td); r /= Td;
  const int a = (int)(r % Ad);
  const int b = (int)(r / Ad);
  const int h = axis == 0 ? t : a;
  const int w = axis == 0 ? a : t;
  Xa[i] = x[(((long long)b * C + c) * H + h) * W + w];
}

// cat[:, 0:C] = relu(Oh + Ow) scattered back to (B, catC, H, W)
__global__ __launch_bounds__(TPB) void axial_combine_relu(
    const float* __restrict__ Oh, const float* __restrict__ Ow,
    float* __restrict__ cat, int Bv, int C, int H, int W, int catC,
    long long total)
{
  long long i = (long long)blockIdx.x * TPB + threadIdx.x;
  if (i >= total) return;
  const int w = (int)(i % W);  long long r = i / W;
  const int h = (int)(r % H);  r /= H;
  const int c = (int)(r % C);
  const int b = (int)(r / C);
  const long long rowH = ((long long)b * W + w) * H + h;
  const long long rowW = ((long long)b * H + h) * W + w;
  const float v = Oh[rowH * C + c] + Ow[rowW * C + c];
  cat[(((long long)b * catC + c) * H + h) * W + w] = fmaxf(v, 0.f);
}

// cat[:, coff:coff+C] = src (B,C,H,W)
__global__ __launch_bounds__(TPB) void copy_channels(
    const float* __restrict__ src, float* __restrict__ cat,
    int Bv, int C, int H, int W, int catC, int coff, long long total)
{
  long long i = (long long)blockIdx.x * TPB + threadIdx.x;
  if (i >= total) return;
  const int w = (int)(i % W);  long long r = i / W;
  const int h = (int)(r % H);  r /= H;
  const int c = (int)(r % C);
  const int b = (int)(r / C);
  cat[(((long long)b * catC + (coff + c)) * H + h) * W + w] = src[i];
}

// 2x2 max-pool; H,W are INPUT dims.
__global__ __launch_bounds__(TPB) void maxpool2(
    const float* __restrict__ x, float* __restrict__ y,
    int Bv, int C, int H, int W, long long total)
{
  long long i = (long long)blockIdx.x * TPB + threadIdx.x;
  if (i >= total) return;
  const int Wo = W >> 1, Ho = H >> 1;
  const int ow = (int)(i % Wo);  long long r = i / Wo;
  const int oh = (int)(r % Ho);  r /= Ho;
  const int c  = (int)(r % C);
  const int b  = (int)(r / C);
  const float* p = x + (((long long)b * C + c) * H + 2 * oh) * W + 2 * ow;
  y[i] = fmaxf(fmaxf(p[0], p[1]), fmaxf(p[W], p[W + 1]));
}

// bilinear x2 upsample, align_corners=True; H,W are INPUT dims.
__global__ __launch_bounds__(TPB) void upsample2(
    const float* __restrict__ x, float* __restrict__ y,
    int Bv, int C, int H, int W, long long total)
{
  long long i = (long long)blockIdx.x * TPB + threadIdx.x;
  if (i >= total) return;
  const int Wo = 2 * W, Ho = 2 * H;
  const int ow = (int)(i % Wo);  long long r = i / Wo;
  const int oh = (int)(r % Ho);  r /= Ho;
  const int c  = (int)(r % C);
  const int b  = (int)(r / C);
  const float sh = oh * ((float)(H - 1) / (float)(Ho - 1));
  const float sw = ow * ((float)(W - 1) / (float)(Wo - 1));
  int h0 = (int)floorf(sh); int h1 = min(h0 + 1, H - 1); float fh = sh - (float)h0;
  int w0 = (int)floorf(sw); int w1 = min(w0 + 1, W - 1); float fw = sw - (float)w0;
  const float* p = x + (((long long)b * C + c) * H) * W;
  const float v0 = p[(long long)h0 * W + w0] * (1.f - fw) + p[(long long)h0 * W + w1] * fw;
  const float v1 = p[(long long)h1 * W + w0] * (1.f - fw) + p[(long long)h1 * W + w1] * fw;
  y[i] = v0 * (1.f - fh) + v1 * fh;
}

// ---------------- BatchNorm (training-mode batch stats) --------------------
__global__ __launch_bounds__(TPB) void bn_zero(float* __restrict__ s, int n)
{
  int i = blockIdx.x * TPB + threadIdx.x;
  if (i < n) s[i] = 0.f;
}

__global__ __launch_bounds__(TPB) void bn_stats(
    const float* __restrict__ x, float* __restrict__ sums,
    int Bv, int C, int H, int W)
{
  const int c = blockIdx.y;
  const long long HWn = (long long)H * W;
  const long long n = (long long)Bv * HWn;
  float s = 0.f, ss = 0.f;
  for (long long i = (long long)blockIdx.x * TPB + threadIdx.x; i < n;
       i += (long long)gridDim.x * TPB) {
    const int b = (int)(i / HWn);
    const long long p = i % HWn;
    const float v = x[((long long)b * C + c) * HWn + p];
    s += v; ss += v * v;
  }
  for (int o = 16; o; o >>= 1) { s += __shfl_xor(s, o, 32); ss += __shfl_xor(ss, o, 32); }
  __shared__ float sh0[TPB / 32], sh1[TPB / 32];
  if ((threadIdx.x & 31) == 0) { sh0[threadIdx.x >> 5] = s; sh1[threadIdx.x >> 5] = ss; }
  __syncthreads();
  if (threadIdx.x == 0) {
    float ts = 0.f, tss = 0.f;
    for (int v = 0; v < TPB / 32; ++v) { ts += sh0[v]; tss += sh1[v]; }
    atomicAdd(&sums[c], ts);
    atomicAdd(&sums[C + c], tss);
  }
}

__global__ __launch_bounds__(TPB) void bn_apply(
    float* __restrict__ x, const float* __restrict__ sums,
    const float* __restrict__ g, const float* __restrict__ bta,
    int Bv, int C, int H, int W, long long total)
{
  long long i = (long long)blockIdx.x * TPB + threadIdx.x;
  if (i >= total) return;
  const long long HWn = (long long)H * W;
  const int c = (int)((i / HWn) % C);
  const float cnt = (float)((long long)Bv * HWn);
  const float mean = sums[c] / cnt;
  const float var  = sums[C + c] / cnt - mean * mean;
  const float sc   = g[c] * rsqrtf(var + 1e-5f);
  x[i] = (x[i] - mean) * sc + bta[c];
}

// ===========================================================================
// Host-side orchestration
// ===========================================================================
static inline unsigned nblk(long long n) { return (unsigned)((n + TPB - 1) / TPB); }

static void launch_gemm(hipStream_t s, const float* A, const float* B, float* C,
                        const float* bias, int M, int N, int K,
                        int lda, int ldb, int ldc,
                        long long sA, long long sB, long long sC,
                        int batches, float alpha, int transB, int relu)
{
  const int tilesM  = (M + 15) / 16;
  const int tilesN4 = (N + 63) / 64;
  const long long tiles = (long long)tilesM * tilesN4;
  dim3 grid((unsigned)((tiles + (TPB / 32) - 1) / (TPB / 32)), (unsigned)batches);
  if (transB)
    gemm_wmma_f32<1><<<grid, TPB, 0, s>>>(A, B, C, bias, M, N, K, lda, ldb, ldc,
                                          sA, sB, sC, alpha, relu, tilesN4);
  else
    gemm_wmma_f32<0><<<grid, TPB, 0, s>>>(A, B, C, bias, M, N, K, lda, ldb, ldc,
                                          sA, sB, sC, alpha, relu, tilesN4);
}

// Self-attention along one axis (lucidrains SelfAttention, HEADS=2).
static void run_axis_attn(hipStream_t s, const float* xin,
                          int Bv, int C, int H, int W, int axis,
                          const float* Wq, const float* Wkv, const float* Wo,
                          const float* bo, float* Oax, float* arena)
{
  const int Ad = (axis == 0) ? W : H;
  const int Td = (axis == 0) ? H : W;
  const long long tokens = (long long)Bv * Ad * Td;
  const int hid = C, e = C / 2;     // hid = (C/HEADS)*HEADS = C (C even), e = C/2
  const int BA = Bv * Ad;

  float* Xa = arena;
  float* Q  = Xa + tokens * C;
  float* KV = Q + tokens * hid;
  float* AO = KV + tokens * 2LL * hid;
  float* DT = AO + tokens * hid;    // (HEADS, BA, Td, Td)

  pack_axis<<<nblk(tokens * C), TPB, 0, s>>>(xin, Xa, Bv, C, H, W, axis, tokens * C);

  // projections
  launch_gemm(s, Xa, Wq,  Q,  nullptr, (int)tokens, hid,     C, C, hid,     hid,     0, 0, 0, 1, 1.f, 0, 0);
  launch_gemm(s, Xa, Wkv, KV, nullptr, (int)tokens, 2 * hid, C, C, 2 * hid, 2 * hid, 0, 0, 0, 1, 1.f, 0, 0);

  const float scale = 1.0f / sqrtf((float)e);
  for (int hd = 0; hd < 2; ++hd) {
    // dots = Q_h (Td x e) * K_h (Td x e)^T, batched over BA
    launch_gemm(s, Q + hd * e, KV + hd * e, DT + (long long)hd * BA * Td * Td,
                nullptr, Td, Td, e, hid, 2 * hid, Td,
                (long long)Td * hid, (long long)Td * 2 * hid, (long long)Td * Td,
                BA, scale, /*transB=*/1, 0);
  }
  softmax_rows<<<nblk((2LL * BA * Td) * 32), TPB, 0, s>>>(DT, 2LL * BA * Td, Td);
  for (int hd = 0; hd < 2; ++hd) {
    // out_h = attn (Td x Td) * V_h (Td x e)
    launch_gemm(s, DT + (long long)hd * BA * Td * Td, KV + hid + hd * e, AO + hd * e,
                nullptr, Td, e, Td, Td, 2 * hid, hid,
                (long long)Td * Td, (long long)Td * 2 * hid, (long long)Td * hid,
                BA, 1.f, 0, 0);
  }
  // Oax = AO @ Wo + bo
  launch_gemm(s, AO, Wo, Oax, bo, (int)tokens, C, hid, hid, C, C, 0, 0, 0, 1, 1.f, 0, 0);
}

struct Blk {
  const float *Wq_h, *Wkv_h, *Wo_h, *bo_h, *Wq_w, *Wkv_w, *Wo_w, *bo_w;
  const float *conv_w, *bn_g, *bn_b;
};

static Blk get_blk(void* const* d_in, int base)
{
  Blk b;
  b.Wq_h  = (const float*)d_in[base + 0];
  b.Wkv_h = (const float*)d_in[base + 1];
  b.Wo_h  = (const float*)d_in[base + 2];
  b.bo_h  = (const float*)d_in[base + 3];
  b.Wq_w  = (const float*)d_in[base + 4];
  b.Wkv_w = (const float*)d_in[base + 5];
  b.Wo_w  = (const float*)d_in[base + 6];
  b.bo_w  = (const float*)d_in[base + 7];
  b.conv_w = (const float*)d_in[base + 8];
  b.bn_g   = (const float*)d_in[base + 9];
  b.bn_b   = (const float*)d_in[base + 10];
  return b;
}

// relu(axial_attn(x)) -> cat[a, x, extra] -> relu(conv1x1) -> BN
static void run_block(hipStream_t s, const float* xin, int Bv, int C, int H, int W,
                      const float* extra, int extraC, const Blk& p, int outC,
                      float* out, float* Oh, float* Ow, float* arena, float* bns)
{
  run_axis_attn(s, xin, Bv, C, H, W, 0, p.Wq_h, p.Wkv_h, p.Wo_h, p.bo_h, Oh, arena);
  run_axis_attn(s, xin, Bv, C, H, W, 1, p.Wq_w, p.Wkv_w, p.Wo_w, p.bo_w, Ow, arena);

  const int catC = 2 * C + extraC;
  const long long HWn = (long long)H * W;
  float* cat = arena;   // attention temps are dead now

  axial_combine_relu<<<nblk((long long)Bv * C * HWn), TPB, 0, s>>>(
      Oh, Ow, cat, Bv, C, H, W, catC, (long long)Bv * C * HWn);
  copy_channels<<<nblk((long long)Bv * C * HWn), TPB, 0, s>>>(
      xin, cat, Bv, C, H, W, catC, C, (long long)Bv * C * HWn);
  if (extra)
    copy_channels<<<nblk((long long)Bv * extraC * HWn), TPB, 0, s>>>(
        extra, cat, Bv, extraC, H, W, catC, 2 * C, (long long)Bv * extraC * HWn);

  // conv1x1 + relu: per-batch GEMM (outC x catC) @ (catC x HW)
  launch_gemm(s, p.conv_w, cat, out, nullptr, outC, (int)HWn, catC,
              catC, (int)HWn, (int)HWn,
              0, (long long)catC * HWn, (long long)outC * HWn, Bv, 1.f, 0, /*relu=*/1);

  // BatchNorm (training-mode statistics)
  bn_zero<<<nblk(2 * outC), TPB, 0, s>>>(bns, 2 * outC);
  {
    dim3 grid(64, (unsigned)outC);
    bn_stats<<<grid, TPB, 0, s>>>(out, bns, Bv, outC, H, W);
  }
  bn_apply<<<nblk((long long)Bv * outC * HWn), TPB, 0, s>>>(
      out, bns, p.bn_g, p.bn_b, Bv, outC, H, W, (long long)Bv * outC * HWn);
}

extern "C" void kernel_launch(void* const* d_in, const int* in_sizes, int n_in,
                              void* d_out, int out_size, void* d_ws, size_t ws_size,
                              hipStream_t stream)
{
  // Input order assumption: setup_inputs() insertion order, tree-flattened:
  //   0: x (2,4,256,256)
  //   blocks encode,down1..4,up1..4 at bases 1,12,23,34,45,56,67,78,89
  //     each: Wq_h,Wkv_h,Wo_h,bo_h, Wq_w,Wkv_w,Wo_w,bo_w, conv_w,bn_g,bn_b
  //   100: decode_w (2,64)
  (void)in_sizes; (void)n_in; (void)out_size; (void)ws_size;

  const float* x = (const float*)d_in[0];
  const float* decode_w = (const float*)d_in[100];
  const Blk enc = get_blk(d_in, 1);
  const Blk dn1 = get_blk(d_in, 12), dn2 = get_blk(d_in, 23);
  const Blk dn3 = get_blk(d_in, 34), dn4 = get_blk(d_in, 45);
  const Blk up1 = get_blk(d_in, 56), up2 = get_blk(d_in, 67);
  const Blk up3 = get_blk(d_in, 78), up4 = get_blk(d_in, 89);

  // ---------------- workspace layout (floats) ----------------
  float* ws = (float*)d_ws;
  size_t o = 0;
  float* X1   = ws + o; o += 8388608ull;    // 64  @ 2x256x256
  float* X2   = ws + o; o += 4194304ull;    // 128 @ 2x128x128
  float* X3   = ws + o; o += 2097152ull;    // 256 @ 2x64x64
  float* X4   = ws + o; o += 1048576ull;    // 512 @ 2x32x32
  float* CURA = ws + o; o += 16777216ull;   // block inputs (pool/upsample)
  float* CURB = ws + o; o += 8388608ull;    // block outputs (x5..x9)
  float* OH   = ws + o; o += 16777216ull;   // h-axis attention output
  float* OW   = ws + o; o += 16777216ull;   // w-axis attention output
  float* ARENA= ws + o; o += 150994944ull;  // attn temps / cat (union)
  float* BNS  = ws + o; o += 2048ull;       // BN sum/sumsq

  const int Bv = 2;
  hipStream_t s = stream;

  // encode @ 256: dim 4 -> 64
  run_block(s, x, Bv, 4, 256, 256, nullptr, 0, enc, 64, X1, OH, OW, ARENA, BNS);
  // down1: pool -> 64 @128 -> 128
  maxpool2<<<nblk(2LL * 64 * 128 * 128), TPB, 0, s>>>(X1, CURA, Bv, 64, 256, 256, 2LL * 64 * 128 * 128);
  run_block(s, CURA, Bv, 64, 128, 128, nullptr, 0, dn1, 128, X2, OH, OW, ARENA, BNS);
  // down2
  maxpool2<<<nblk(2LL * 128 * 64 * 64), TPB, 0, s>>>(X2, CURA, Bv, 128, 128, 128, 2LL * 128 * 64 * 64);
  run_block(s, CURA, Bv, 128, 64, 64, nullptr, 0, dn2, 256, X3, OH, OW, ARENA, BNS);
  // down3
  maxpool2<<<nblk(2LL * 256 * 32 * 32), TPB, 0, s>>>(X3, CURA, Bv, 256, 64, 64, 2LL * 256 * 32 * 32);
  run_block(s, CURA, Bv, 256, 32, 32, nullptr, 0, dn3, 512, X4, OH, OW, ARENA, BNS);
  // down4 -> x5 in CURB
  maxpool2<<<nblk(2LL * 512 * 16 * 16), TPB, 0, s>>>(X4, CURA, Bv, 512, 32, 32, 2LL * 512 * 16 * 16);
  run_block(s, CURA, Bv, 512, 16, 16, nullptr, 0, dn4, 1024, CURB, OH, OW, ARENA, BNS);
  // up1: upsample x5 -> 1024 @32, cat x4 -> 512 (x6)
  upsample2<<<nblk(2LL * 1024 * 32 * 32), TPB, 0, s>>>(CURB, CURA, Bv, 1024, 16, 16, 2LL * 1024 * 32 * 32);
  run_block(s, CURA, Bv, 1024, 32, 32, X4, 512, up1, 512, CURB, OH, OW, ARENA, BNS);
  // up2 (x7)
  upsample2<<<nblk(2LL * 512 * 64 * 64), TPB, 0, s>>>(CURB, CURA, Bv, 512, 32, 32, 2LL * 512 * 64 * 64);
  run_block(s, CURA, Bv, 512, 64, 64, X3, 256, up2, 256, CURB, OH, OW, ARENA, BNS);
  // up3 (x8)
  upsample2<<<nblk(2LL * 256 * 128 * 128), TPB, 0, s>>>(CURB, CURA, Bv, 256, 64, 64, 2LL * 256 * 128 * 128);
  run_block(s, CURA, Bv, 256, 128, 128, X2, 128, up3, 128, CURB, OH, OW, ARENA, BNS);
  // up4 (x9)
  upsample2<<<nblk(2LL * 128 * 256 * 256), TPB, 0, s>>>(CURB, CURA, Bv, 128, 128, 128, 2LL * 128 * 256 * 256);
  run_block(s, CURA, Bv, 128, 256, 256, X1, 64, up4, 64, CURB, OH, OW, ARENA, BNS);

  // decode: (2 x 64) @ (64 x 65536) per batch -> d_out (2,2,256,256)
  const long long HW0 = 256LL * 256LL;
  launch_gemm(s, decode_w, CURB, (float*)d_out, nullptr,
              2, (int)HW0, 64, 64, (int)HW0, (int)HW0,
              0, 64LL * HW0, 2LL * HW0, Bv, 1.f, 0, 0);
}